// CNN2D_bilinearAttPool_171798692279
// MI455X (gfx1250) — compile-verified
//
#include <hip/hip_runtime.h>
#include <math.h>

#define EPSV 1e-5f
#define B_   16
#define C_   512
#define N_   3136   // 56*56
#define NBVEC 3

typedef __attribute__((ext_vector_type(2))) float v2f;
typedef __attribute__((ext_vector_type(8))) float v8f;

// ---------------------------------------------------------------------------
// K0: norm[b,n] = sqrt(sum_c x^2) + EPS ; score = norm  (one thread per pos)
// ---------------------------------------------------------------------------
__global__ void k_norm(const float* __restrict__ X, float* __restrict__ norm,
                       float* __restrict__ score) {
    int b = blockIdx.y;
    int n = blockIdx.x * blockDim.x + threadIdx.x;
    if (n >= N_) return;
    const float* xb = X + (size_t)b * C_ * N_ + n;
    float acc = 0.f;
#pragma unroll 8
    for (int c = 0; c < C_; ++c) {
        float t = xb[(size_t)c * N_];
        acc += t * t;
    }
    float nv = sqrtf(acc) + EPSV;
    norm[b * N_ + n]  = nv;
    score[b * N_ + n] = nv;
}

// ---------------------------------------------------------------------------
// K1: per-batch argmax of score (first-max tie-break, matching jnp.argmax),
//     then gather v[b,c] = x[b,c,ind], vnorm[b] = norm[b,ind]
// ---------------------------------------------------------------------------
__global__ void k_argmax_gather(const float* __restrict__ X,
                                const float* __restrict__ norm,
                                const float* __restrict__ score,
                                float* __restrict__ v, float* __restrict__ vnorm) {
    __shared__ float sval[256];
    __shared__ int   sidx[256];
    int b = blockIdx.x, tid = threadIdx.x;
    float best = -3.0e38f; int bi = 0;
    for (int n = tid; n < N_; n += 256) {     // ascending n: strict > keeps first max
        float s = score[b * N_ + n];
        if (s > best) { best = s; bi = n; }
    }
    sval[tid] = best; sidx[tid] = bi;
    __syncthreads();
    for (int off = 128; off > 0; off >>= 1) {
        if (tid < off) {
            float ov = sval[tid + off]; int oi = sidx[tid + off];
            if (ov > sval[tid] || (ov == sval[tid] && oi < sidx[tid])) {
                sval[tid] = ov; sidx[tid] = oi;
            }
        }
        __syncthreads();
    }
    int ind = sidx[0];
    for (int c = tid; c < C_; c += 256)
        v[b * C_ + c] = X[((size_t)(b * C_ + c)) * N_ + ind];
    if (tid == 0) vnorm[b] = norm[b * N_ + ind];
}

// ---------------------------------------------------------------------------
// K2: sim GEMV via V_WMMA_F32_16X16X4_F32.
//     One wave -> 16 consecutive positions. A[16x4] = x tile (positions x
//     4-channel chunk); B[4x16] = gathered vector chunk broadcast across all
//     16 columns, so every D column equals X·v. 128 WMMAs sweep C=512.
//     Epilogue (lanes 0/16): sim = dot/(norm*vnorm), score *= (1-sim),
//     block partial sums of sim -> partials.
//     blockDim = 128 (4 waves = 64 positions); grid.x = 49 -> 49*64 = 3136 = N.
// ---------------------------------------------------------------------------
__global__ void k_sim_wmma(const float* __restrict__ X, const float* __restrict__ vvec,
                           const float* __restrict__ vnorm, const float* __restrict__ norm,
                           float* __restrict__ sim, float* __restrict__ score,
                           float* __restrict__ partials) {
    __shared__ float vl[C_];
    __shared__ float psum[8];
    int b   = blockIdx.y;
    int tid = threadIdx.x;                       // 0..127
    for (int i = tid; i < C_; i += 128) vl[i] = vvec[b * C_ + i];
    if (tid < 8) psum[tid] = 0.f;
    __syncthreads();

    int wave = tid >> 5;                         // 0..3
    int lane = tid & 31;
    int n0   = blockIdx.x * 64 + wave * 16;      // wave's 16 positions
    int m    = lane & 15;                        // A-matrix row (position)
    int g    = lane >> 4;                        // lane group
    int kb   = g * 2;                            // K base held by this lane

    const float* xa = X + (size_t)b * C_ * N_ + n0 + m;
    v8f acc = {};
#pragma unroll 4
    for (int c0 = 0; c0 < C_; c0 += 4) {
        v2f a, bm;
        // A 16x4 f32 layout: lane m holds K=kb (v0), K=kb+1 (v1)
        a.x  = xa[(size_t)(c0 + kb)     * N_];
        a.y  = xa[(size_t)(c0 + kb + 1) * N_];
        // B 4x16: row K constant across N (broadcast of v)
        bm.x = vl[c0 + kb];
        bm.y = vl[c0 + kb + 1];
        acc = __builtin_amdgcn_wmma_f32_16x16x4_f32(
            /*neg_a=*/false, a, /*neg_b=*/false, bm,
            /*c_mod=*/(short)0, acc, /*reuse_a=*/false, /*reuse_b=*/false);
    }

    // D layout: VGPR r, lanes 0-15 -> M=r; lanes 16-31 -> M=r+8 (all cols equal)
    if (m == 0) {
        float vn = vnorm[b];
        float ws = 0.f;
#pragma unroll
        for (int r = 0; r < 8; ++r) {
            int n   = n0 + 8 * g + r;
            float s = acc[r] / (norm[b * N_ + n] * vn);
            sim[b * N_ + n]    = s;
            score[b * N_ + n] *= (1.0f - s);     // suppression uses raw sim
            ws += s;
        }
        psum[wave * 2 + g] = ws;
    }
    __syncthreads();
    if (tid == 0) {
        float t = 0.f;
#pragma unroll
        for (int i = 0; i < 8; ++i) t += psum[i];
        partials[b * 49 + blockIdx.x] = t;
    }
}

// ---------------------------------------------------------------------------
// K3: sumSim[b] = sum of 49 block partials (deterministic tree reduce)
// ---------------------------------------------------------------------------
__global__ void k_sumsim(const float* __restrict__ partials, float* __restrict__ sumSim) {
    __shared__ float s[64];
    int b = blockIdx.x, tid = threadIdx.x;       // 64 threads
    float a = 0.f;
    for (int i = tid; i < 49; i += 64) a += partials[b * 49 + i];
    s[tid] = a; __syncthreads();
    for (int off = 32; off > 0; off >>= 1) {
        if (tid < off) s[tid] += s[tid + off];
        __syncthreads();
    }
    if (tid == 0) sumSim[b] = s[0];
}

// ---------------------------------------------------------------------------
// K4: reprVec[b,t,c] = (sum_n x[b,c,n] * sim[b,n]) / sumSim[b]
//     One wave per (b,c); coalesced row sweep + wave32 butterfly reduce.
// ---------------------------------------------------------------------------
__global__ void k_repr(const float* __restrict__ X, const float* __restrict__ sim,
                       const float* __restrict__ sumSim, float* __restrict__ outRepr,
                       int t) {
    int w    = blockIdx.x * (blockDim.x >> 5) + (threadIdx.x >> 5);
    int lane = threadIdx.x & 31;
    int b = w >> 9;          // / 512
    int c = w & 511;
    const float* xc = X   + ((size_t)(b * C_ + c)) * N_;
    const float* sb = sim + b * N_;
    float acc = 0.f;
    for (int n = lane; n < N_; n += 32) {
        __builtin_prefetch(xc + n + 1024, 0, 1);   // global_prefetch_b8, ~32 iters ahead
        acc += xc[n] * sb[n];
    }
#pragma unroll
    for (int off = 16; off > 0; off >>= 1) acc += __shfl_xor(acc, off, 32);
    if (lane == 0) outRepr[(b * NBVEC + t) * C_ + c] = acc / sumSim[b];
}

// ---------------------------------------------------------------------------
// K5: sims output: out[b,t,n] = sim[b,n] / sumSim[b]
// ---------------------------------------------------------------------------
__global__ void k_simout(const float* __restrict__ sim, const float* __restrict__ sumSim,
                         float* __restrict__ outSims, int t) {
    int b = blockIdx.y;
    int n = blockIdx.x * 256 + threadIdx.x;
    if (n >= N_) return;
    outSims[(b * NBVEC + t) * N_ + n] = sim[b * N_ + n] / sumSim[b];
}

// ---------------------------------------------------------------------------
extern "C" void kernel_launch(void* const* d_in, const int* in_sizes, int n_in,
                              void* d_out, int out_size, void* d_ws, size_t ws_size,
                              hipStream_t stream) {
    const float* X = (const float*)d_in[0];
    // d_in[1] = nbVec (scalar 3); loop count is static below.

    float* ws       = (float*)d_ws;
    float* norm     = ws;                       // 16*3136
    float* score    = ws + 50176;               // 16*3136
    float* sim      = ws + 100352;              // 16*3136
    float* v        = ws + 150528;              // 16*512
    float* vnorm    = ws + 158720;              // 16
    float* sumSim   = ws + 158736;              // 16
    float* partials = ws + 158752;              // 16*49

    float* outRepr = (float*)d_out;                       // [16,3,512]
    float* outSims = (float*)d_out + B_ * NBVEC * C_;     // [16,3,56,56]

    k_norm<<<dim3(13, 16), 256, 0, stream>>>(X, norm, score);
    for (int t = 0; t < NBVEC; ++t) {
        k_argmax_gather<<<16, 256, 0, stream>>>(X, norm, score, v, vnorm);
        k_sim_wmma<<<dim3(49, 16), 128, 0, stream>>>(X, v, vnorm, norm, sim, score, partials);
        k_sumsim<<<16, 64, 0, stream>>>(partials, sumSim);
        k_repr<<<1024, 256, 0, stream>>>(X, sim, sumSim, outRepr, t);
        k_simout<<<dim3(13, 16), 256, 0, stream>>>(sim, sumSim, outSims, t);
    }
}